// AssociativeSparseDistributedMemory_52587579572260
// MI455X (gfx1250) — compile-verified
//
#include <hip/hip_runtime.h>
#include <hip/hip_bf16.h>
#include <math.h>

typedef __attribute__((ext_vector_type(2))) float v2f;
typedef __attribute__((ext_vector_type(8))) float v8f;

#define B_ROWS   128
#define M_TILES  8          // 128 rows / 16
#define NEG_INF  (-3.402823466e38f)

// ---------------------------------------------------------------------------
// scores[M=128, N] = A[128, K] @ W[N, K]^T   via V_WMMA_F32_16X16X4_F32
// block = 128 threads = 4 waves; each wave owns one 16-column N tile and all
// 8 M tiles (B fragment reused 8x, W streamed from HBM exactly once).
// ---------------------------------------------------------------------------
__global__ __launch_bounds__(128) void gemm_nt_f32_wmma(
    const float* __restrict__ A, const float* __restrict__ W,
    float* __restrict__ out, int N, int K) {
  const int wave  = threadIdx.x >> 5;           // 0..3
  const int lane  = threadIdx.x & 31;
  const int ntile = blockIdx.x * 4 + wave;      // 16-wide column tile
  const int r     = lane & 15;                  // row-within-tile / col-within-tile
  const int ks    = (lane >> 4) * 2;            // f32 A/B frag K sub-offset

  const float* wrow = W + (size_t)(ntile * 16 + r) * K;

  v8f acc[M_TILES];
#pragma unroll
  for (int t = 0; t < M_TILES; ++t) acc[t] = (v8f)0.0f;

  for (int k0 = 0; k0 < K; k0 += 4) {
    if ((k0 & 63) == 0) __builtin_prefetch(wrow + k0 + 256, 0, 0);
    v2f b = *(const v2f*)(wrow + k0 + ks);
#pragma unroll
    for (int t = 0; t < M_TILES; ++t) {
      v2f a = *(const v2f*)(A + (size_t)(t * 16 + r) * K + k0 + ks);
      acc[t] = __builtin_amdgcn_wmma_f32_16x16x4_f32(
          /*neg_a=*/false, a, /*neg_b=*/false, b,
          /*c_mod=*/(short)0, acc[t], /*reuse_a=*/false, /*reuse_b=*/false);
    }
  }

  // C/D layout: VGPR v, lanes 0-15 -> M=v, lanes 16-31 -> M=v+8; N = lane&15
  const int col  = ntile * 16 + (lane & 15);
  const int rofs = (lane >> 4) * 8;
#pragma unroll
  for (int t = 0; t < M_TILES; ++t) {
#pragma unroll
    for (int v = 0; v < 8; ++v) {
      const int row = t * 16 + rofs + v;
      out[(size_t)row * N + col] = acc[t][v];
    }
  }
}

// ---------------------------------------------------------------------------
// Per-row top-32 argmax. One workgroup per row; whole row staged in LDS
// (CDNA5: 320 KB/WGP, a 32768-float row = 128 KB fits), then 32 argmax
// passes with a tree reduction — HBM touched exactly once.
// ---------------------------------------------------------------------------
__global__ __launch_bounds__(1024) void topk32_kernel(
    const float* __restrict__ scores, int n, int* __restrict__ idx_out) {
  extern __shared__ float smem[];
  float* s_row = smem;                 // n floats
  float* s_val = smem + n;             // 1024 floats
  int*   s_idx = (int*)(s_val + 1024); // 1024 ints

  const int b = blockIdx.x;
  const float* row = scores + (size_t)b * n;
  for (int i = threadIdx.x; i < n; i += 1024) s_row[i] = row[i];
  __syncthreads();

  for (int j = 0; j < 32; ++j) {
    float best = NEG_INF; int bi = 0;
    for (int i = threadIdx.x; i < n; i += 1024) {
      float v = s_row[i];
      if (v > best) { best = v; bi = i; }
    }
    s_val[threadIdx.x] = best; s_idx[threadIdx.x] = bi;
    __syncthreads();
    for (int off = 512; off > 0; off >>= 1) {
      if (threadIdx.x < off) {
        float o = s_val[threadIdx.x + off];
        if (o > s_val[threadIdx.x]) {
          s_val[threadIdx.x] = o;
          s_idx[threadIdx.x] = s_idx[threadIdx.x + off];
        }
      }
      __syncthreads();
    }
    if (threadIdx.x == 0) {
      idx_out[b * 32 + j] = s_idx[0];
      s_row[s_idx[0]] = NEG_INF;       // exclude from later passes
    }
    __syncthreads();
  }
}

// ---------------------------------------------------------------------------
// p[b,:] = l2normalize( (1/sqrt(32)) * sum_j enc[idx[b,j], :] ),  PROJ=4096
// ---------------------------------------------------------------------------
__global__ __launch_bounds__(256) void gather_sum_norm_kernel(
    const int* __restrict__ idx, const float* __restrict__ enc,
    float* __restrict__ p) {
  __shared__ int   s_idx[32];
  __shared__ float s_red[256];
  const int b = blockIdx.x;
  if (threadIdx.x < 32) s_idx[threadIdx.x] = idx[b * 32 + threadIdx.x];
  __syncthreads();

  float acc[16];
#pragma unroll
  for (int i = 0; i < 16; ++i) acc[i] = 0.0f;

  for (int j = 0; j < 32; ++j) {
    const float* r = enc + (size_t)s_idx[j] * 4096;
    if (j < 31) __builtin_prefetch(enc + (size_t)s_idx[j + 1] * 4096 + threadIdx.x, 0, 0);
#pragma unroll
    for (int i = 0; i < 16; ++i) acc[i] += r[threadIdx.x + i * 256];
  }

  const float scale = 0.17677669529663687f;  // 1/sqrt(32)
  float ss = 0.0f;
#pragma unroll
  for (int i = 0; i < 16; ++i) { acc[i] *= scale; ss += acc[i] * acc[i]; }

  s_red[threadIdx.x] = ss;
  __syncthreads();
  for (int off = 128; off > 0; off >>= 1) {
    if (threadIdx.x < off) s_red[threadIdx.x] += s_red[threadIdx.x + off];
    __syncthreads();
  }
  const float inv = 1.0f / fmaxf(sqrtf(s_red[0]), 1e-12f);
#pragma unroll
  for (int i = 0; i < 16; ++i)
    p[(size_t)b * 4096 + threadIdx.x + i * 256] = acc[i] * inv;
}

// ---------------------------------------------------------------------------
// out[b,:] = sum_j mem[idx[b,j], :],  VALUE_CAP = 32768 (float4 coalesced)
// ---------------------------------------------------------------------------
__global__ __launch_bounds__(256) void gather_sum_out_kernel(
    const int* __restrict__ idx, const float* __restrict__ mem,
    float* __restrict__ out) {
  __shared__ int s_idx[32];
  const int b = blockIdx.y;
  if (threadIdx.x < 32) s_idx[threadIdx.x] = idx[b * 32 + threadIdx.x];
  __syncthreads();

  const int col = (blockIdx.x * 256 + threadIdx.x) * 4;
  float4 acc = make_float4(0.f, 0.f, 0.f, 0.f);
  for (int j = 0; j < 32; ++j) {
    if (j < 31) __builtin_prefetch(mem + (size_t)s_idx[j + 1] * 32768 + col, 0, 0);
    const float4 v = *(const float4*)(mem + (size_t)s_idx[j] * 32768 + col);
    acc.x += v.x; acc.y += v.y; acc.z += v.z; acc.w += v.w;
  }
  *(float4*)(out + (size_t)b * 32768 + col) = acc;
}

// ---------------------------------------------------------------------------
extern "C" void kernel_launch(void* const* d_in, const int* in_sizes, int n_in,
                              void* d_out, int out_size, void* d_ws, size_t ws_size,
                              hipStream_t stream) {
  (void)in_sizes; (void)n_in; (void)out_size; (void)ws_size;
  const float* keys        = (const float*)d_in[0];  // [128, 1024]
  const float* value_proj  = (const float*)d_in[1];  // [32768, 1024]
  const float* cliq_enc    = (const float*)d_in[2];  // [32768, 4096]
  const float* assoc_proj  = (const float*)d_in[3];  // [4096, 4096]
  const float* assoc_mem   = (const float*)d_in[4];  // [4096, 32768]
  float* out = (float*)d_out;                        // [128, 32768]

  char* ws = (char*)d_ws;
  float* scores1   = (float*)(ws);                               // 128*32768 f32
  int*   cliq_idx  = (int*)  (ws + (size_t)16777216);            // 128*32 i32
  float* p         = (float*)(ws + (size_t)16793600);            // 128*4096 f32
  float* scores2   = (float*)(ws + (size_t)18890752);            // 128*4096 f32
  int*   assoc_idx = (int*)  (ws + (size_t)20987904);            // 128*32 i32

  // 1) scores1 = keys @ value_proj^T     [128, 32768]
  gemm_nt_f32_wmma<<<32768 / 64, 128, 0, stream>>>(keys, value_proj, scores1,
                                                   32768, 1024);
  // 2) top-32 per row (row staged in 128 KB of LDS)
  topk32_kernel<<<B_ROWS, 1024, 32768 * 4 + 1024 * 8, stream>>>(scores1, 32768,
                                                                cliq_idx);
  // 3) p = normalize(sum of gathered encoder rows / sqrt(32))
  gather_sum_norm_kernel<<<B_ROWS, 256, 0, stream>>>(cliq_idx, cliq_enc, p);
  // 4) scores2 = p @ assoc_proj^T        [128, 4096]
  gemm_nt_f32_wmma<<<4096 / 64, 128, 0, stream>>>(p, assoc_proj, scores2,
                                                  4096, 4096);
  // 5) top-32 per row
  topk32_kernel<<<B_ROWS, 1024, 4096 * 4 + 1024 * 8, stream>>>(scores2, 4096,
                                                               assoc_idx);
  // 6) out = sum of gathered assoc_mem_value rows
  gather_sum_out_kernel<<<dim3(32, B_ROWS), 256, 0, stream>>>(assoc_idx,
                                                              assoc_mem, out);
}